// EQGPAgent_25245817766531
// MI455X (gfx1250) — compile-verified
//
#include <hip/hip_runtime.h>
#include <hip/hip_bf16.h>
#include <math.h>
#include <stdint.h>

// ---- problem constants (match reference) ----
#define NN 1024      // transitions
#define GG 32        // state dim / number of GP kernels
#define DD 40        // state+action dim
#define PB 32        // Cholesky panel width
#define NPAN (NN/PB) // 32 panels

typedef __attribute__((ext_vector_type(2))) float v2f;
typedef __attribute__((ext_vector_type(8))) float v8f;
typedef __attribute__((ext_vector_type(4))) uint32_t u32x4;
typedef __attribute__((ext_vector_type(8))) uint32_t u32x8;

// =====================================================================
// TDM helper: issue a tensor_load_to_lds moving a 32x32 f32 tile from a
// row-major matrix (element row stride NN) into an LDS tile laid out as
// [32][33] (the +1 DWORD pad is produced by the TDM pad feature:
// pad_interval code 4 = every 32 DWORDs, pad_amount code 0 = 1 DWORD).
// 2-operand form => VADDR2/VADDR3 = NULL (<=2D tensor). Wave-uniform args.
// =====================================================================
__device__ __forceinline__ void tdm_load_tile32(const float* gsrc, uint32_t lds_byte_off) {
  uint64_t ga = (uint64_t)(uintptr_t)gsrc;
  u32x4 g0;
  g0[0] = 1u;                                   // count=1 (valid), is_restore=0, gather=0
  g0[1] = lds_byte_off;                         // lds_addr
  g0[2] = (uint32_t)ga;                         // global_addr[31:0]
  g0[3] = ((uint32_t)(ga >> 32) & 0x01FFFFFFu)  // global_addr[56:32]
          | 0x80000000u;                        // type=2 ("image") in bits[127:126]
  u32x8 g1;
  g1[0] = (2u << 16)        // data_size = 2 -> 4 bytes
        | (1u << 20)        // pad_enable
        | (4u << 22);       // pad_interval code 4 -> every 32 DWORDs
                            // pad_amount bits[31:25] = 0 -> 1 DWORD
  g1[1] = (uint32_t)(NN & 0xFFFF) << 16;        // tensor_dim0 low16 (=1024)
  g1[2] = ((uint32_t)NN >> 16)                  // tensor_dim0 high16 (=0)
        | ((uint32_t)(NN & 0xFFFF) << 16);      // tensor_dim1 low16 (=1024)
  g1[3] = ((uint32_t)NN >> 16)                  // tensor_dim1 high16 (=0)
        | (32u << 16);                          // tile_dim0 = 32 elements
  g1[4] = 32u;                                  // tile_dim1 = 32 rows; tile_dim2 = 0
  g1[5] = (uint32_t)NN;                         // tensor_dim0_stride low32 = 1024
  g1[6] = 0u;                                   // stride high16 = 0; dim1_stride low16 = 0
  g1[7] = 0u;                                   // dim1_stride high
  asm volatile("tensor_load_to_lds %0, %1" :: "s"(g0), "s"(g1) : "memory");
}

__device__ __forceinline__ uint32_t lds_offset_of(const void* p) {
  return (uint32_t)(size_t)(__attribute__((address_space(3))) const void*)p;
}

// =====================================================================
// Stage 0: build K[g] = coeff_g * exp(-0.5 * sum_d (xi-xj)^2 / s_gd) + noise_g*I
// grid (16,16,32), block 256, 64x64 tile per block
// =====================================================================
__global__ void k_build(const float* __restrict__ X, const float* __restrict__ coeff,
                        const float* __restrict__ scales, const float* __restrict__ noise,
                        float* __restrict__ Kws) {
  const int g   = blockIdx.z;
  const int tr0 = blockIdx.y * 64;
  const int tc0 = blockIdx.x * 64;
  __shared__ float Xr[64][DD];
  __shared__ float Xc[64][DD];
  __shared__ float Wg[DD];
  const int t = threadIdx.x;
  for (int idx = t; idx < 64 * DD; idx += 256) {
    int r = idx / DD, d = idx % DD;
    Xr[r][d] = X[(size_t)(tr0 + r) * DD + d];
    Xc[r][d] = X[(size_t)(tc0 + r) * DD + d];
  }
  if (t < DD) Wg[t] = 1.0f / scales[g * DD + t];
  __syncthreads();

  const float cf = coeff[g];
  const float nz = noise[g];
  const int ty = t >> 4, tx = t & 15;
  const int rr = ty * 4, cc = tx * 4;

  float acc[4][4];
#pragma unroll
  for (int i = 0; i < 4; ++i)
#pragma unroll
    for (int j = 0; j < 4; ++j) acc[i][j] = 0.0f;

  for (int d = 0; d < DD; ++d) {
    const float w = Wg[d];
    float ar[4] = {Xr[rr + 0][d], Xr[rr + 1][d], Xr[rr + 2][d], Xr[rr + 3][d]};
    float br[4] = {Xc[cc + 0][d], Xc[cc + 1][d], Xc[cc + 2][d], Xc[cc + 3][d]};
#pragma unroll
    for (int i = 0; i < 4; ++i)
#pragma unroll
      for (int j = 0; j < 4; ++j) {
        float df = ar[i] - br[j];
        acc[i][j] = fmaf(df * df, w, acc[i][j]);
      }
  }

  float* Kg = Kws + (size_t)g * NN * NN;
#pragma unroll
  for (int i = 0; i < 4; ++i)
#pragma unroll
    for (int j = 0; j < 4; ++j) {
      int R = tr0 + rr + i, C = tc0 + cc + j;
      float v = cf * __expf(-0.5f * acc[i][j]);
      if (R == C) v += nz;
      Kg[(size_t)R * NN + C] = v;
    }
}

// =====================================================================
// Stage 1a: 32x32 diagonal Cholesky (lower) + explicit inverse of L11.
// One wave (32 threads) per g; wave-synchronous.
// =====================================================================
__global__ void k_potf2(float* __restrict__ Kws, float* __restrict__ Linv, int kp) {
  const int g = blockIdx.x;
  const int t = threadIdx.x; // 32 threads
  __shared__ float A[32][33];
  __shared__ float LI[32][33];
  float* Kg = Kws + (size_t)g * NN * NN;
  const int r0 = kp * PB;

  for (int c = 0; c < 32; ++c) A[t][c] = Kg[(size_t)(r0 + t) * NN + r0 + c];
  __syncthreads();

  for (int j = 0; j < 32; ++j) {
    if (t == j) A[j][j] = sqrtf(A[j][j]);
    __syncthreads();
    if (t > j) A[t][j] = A[t][j] / A[j][j];
    __syncthreads();
    if (t > j) {
      float f = A[t][j];
      for (int c = j + 1; c <= t; ++c) A[t][c] -= f * A[c][j];
    }
    __syncthreads();
  }

  // thread t computes column t of L^{-1} (forward substitution), no cross-lane deps
  for (int i = 0; i < 32; ++i) {
    float v;
    if (i < t) v = 0.0f;
    else if (i == t) v = 1.0f / A[t][t];
    else {
      float s = 0.0f;
      for (int j2 = t; j2 < i; ++j2) s += A[i][j2] * LI[j2][t];
      v = -s / A[i][i];
    }
    LI[i][t] = v;
  }
  __syncthreads();

  for (int c = 0; c < 32; ++c) {
    Kg[(size_t)(r0 + t) * NN + r0 + c] = (c <= t) ? A[t][c] : 0.0f;
    Linv[(size_t)g * 1024 + t * 32 + c] = LI[t][c];
  }
}

// =====================================================================
// Stage 1b: TRSM panel  L21 = A21 * L11^{-T}   (GEMM, WMMA f32 16x16x4)
// grid (m_tiles, 32 g), block 128 (4 waves); 32x32 output tile per block
// =====================================================================
__global__ void k_trsm(float* __restrict__ Kws, const float* __restrict__ Linv, int kp) {
  const int g  = blockIdx.y;
  const int bt = blockIdx.x;
  const int r0 = (kp + 1) * PB + bt * 32;
  const int c0 = kp * PB;
  __shared__ float P[32][33];
  __shared__ float LI[32][33];
  float* Kg = Kws + (size_t)g * NN * NN;
  const int t = threadIdx.x;

  for (int idx = t; idx < 1024; idx += 128) {
    int r = idx >> 5, c = idx & 31;
    P[r][c]  = Kg[(size_t)(r0 + r) * NN + c0 + c];
    LI[r][c] = Linv[(size_t)g * 1024 + idx];
  }
  __syncthreads();

  const int w = t >> 5, l = t & 31;
  const int sm = (w >> 1) * 16, sn = (w & 1) * 16;
  const int hi = l >> 4, m = l & 15;

  v8f acc = {0.f, 0.f, 0.f, 0.f, 0.f, 0.f, 0.f, 0.f};
#pragma unroll
  for (int kk = 0; kk < 32; kk += 4) {
    v2f a, b;
    a.x = P[sm + m][kk + 2 * hi];
    a.y = P[sm + m][kk + 2 * hi + 1];
    // B = Linv^T : B[k][n] = Linv[n][k]
    b.x = LI[sn + m][kk + 2 * hi];
    b.y = LI[sn + m][kk + 2 * hi + 1];
    acc = __builtin_amdgcn_wmma_f32_16x16x4_f32(false, a, false, b, (short)0, acc,
                                                false, false);
  }

#pragma unroll
  for (int v = 0; v < 8; ++v) {
    int R = r0 + sm + v + 8 * hi;
    int C = c0 + sn + m;
    Kg[(size_t)R * NN + C] = acc[v];
  }
}

// =====================================================================
// Stage 1c: SYRK trailing update  A[i,j] -= P_i * P_j^T  (i>=j tiles)
// grid (tri_pairs, 32 g), block 128; WMMA f32 16x16x4.
// Tile loads go through the Tensor Data Mover (async DMA into LDS with
// on-the-fly +1-DWORD row padding), issued by wave 0, TENSORcnt-tracked.
// =====================================================================
__global__ void k_syrk(float* __restrict__ Kws, int kp) {
  const int g = blockIdx.y;
  const int p = blockIdx.x;
  // decode triangular pair index p -> (ti,tj), ti>=tj
  int ti = (int)((sqrtf(8.0f * (float)p + 1.0f) - 1.0f) * 0.5f);
  while ((ti + 1) * (ti + 2) / 2 <= p) ++ti;
  while (ti * (ti + 1) / 2 > p) --ti;
  const int tj = p - ti * (ti + 1) / 2;

  const int r0 = (kp + 1) * PB;
  const int c0 = kp * PB;
  __shared__ float Pi[32][33];
  __shared__ float Pj[32][33];
  float* Kg = Kws + (size_t)g * NN * NN;
  const int t = threadIdx.x;

  if ((t >> 5) == 0) { // wave 0 issues both tensor DMAs (EXEC ignored by TDM)
    tdm_load_tile32(Kg + (size_t)(r0 + ti * 32) * NN + c0, lds_offset_of(&Pi[0][0]));
    tdm_load_tile32(Kg + (size_t)(r0 + tj * 32) * NN + c0, lds_offset_of(&Pj[0][0]));
    __builtin_amdgcn_s_wait_tensorcnt(0);
  }
  __syncthreads();

  const int w = t >> 5, l = t & 31;
  const int sm = (w >> 1) * 16, sn = (w & 1) * 16;
  const int hi = l >> 4, m = l & 15;

  v8f acc = {0.f, 0.f, 0.f, 0.f, 0.f, 0.f, 0.f, 0.f};
#pragma unroll
  for (int kk = 0; kk < 32; kk += 4) {
    v2f a, b;
    a.x = Pi[sm + m][kk + 2 * hi];
    a.y = Pi[sm + m][kk + 2 * hi + 1];
    // B = Pj^T : B[k][n] = Pj[n][k]
    b.x = Pj[sn + m][kk + 2 * hi];
    b.y = Pj[sn + m][kk + 2 * hi + 1];
    acc = __builtin_amdgcn_wmma_f32_16x16x4_f32(false, a, false, b, (short)0, acc,
                                                false, false);
  }

#pragma unroll
  for (int v = 0; v < 8; ++v) {
    int R = r0 + ti * 32 + sm + v + 8 * hi;
    int C = r0 + tj * 32 + sn + m;
    size_t idx = (size_t)R * NN + C;
    Kg[idx] = Kg[idx] - acc[v];
  }
}

// =====================================================================
// Stage 2: solve L L^T alpha = y_g  (one block per g, fwd + bwd substitution)
// 32x32 diagonal solves are wave-synchronous on wave 0.
// =====================================================================
__global__ void k_cholsolve(const float* __restrict__ Kws, const float* __restrict__ Y,
                            float* __restrict__ Alpha) {
  const int g = blockIdx.x;
  const int t = threadIdx.x; // 256
  __shared__ float xv[NN];
  __shared__ float part[32][9];
  __shared__ float rhs[32];
  const float* Kg = Kws + (size_t)g * NN * NN;

  for (int i = t; i < NN; i += 256) xv[i] = Y[(size_t)i * GG + g];
  __syncthreads();

  // ---- forward: z = L^{-1} y ----
  for (int jb = 0; jb < NPAN; ++jb) {
    const int rbase = jb * 32;
    const int rr = t >> 3, l8 = t & 7;
    float s = 0.0f;
    for (int c = l8; c < rbase; c += 8) s += Kg[(size_t)(rbase + rr) * NN + c] * xv[c];
    part[rr][l8] = s;
    __syncthreads();
    if (t < 32) {
      float s2 = 0.0f;
      for (int u = 0; u < 8; ++u) s2 += part[t][u];
      rhs[t] = xv[rbase + t] - s2;
    }
    __syncthreads();
    if (t < 32) { // wave-synchronous 32x32 lower solve
      for (int jj = 0; jj < 32; ++jj) {
        if (t == jj) xv[rbase + jj] = rhs[jj] / Kg[(size_t)(rbase + jj) * NN + rbase + jj];
        if (t > jj) rhs[t] -= Kg[(size_t)(rbase + t) * NN + rbase + jj] * xv[rbase + jj];
      }
    }
    __syncthreads();
  }

  // ---- backward: alpha = L^{-T} z ----
  for (int jb = NPAN - 1; jb >= 0; --jb) {
    const int rbase = jb * 32;
    const int rr = t >> 3, l8 = t & 7;
    float s = 0.0f;
    for (int c = (jb + 1) * 32 + l8; c < NN; c += 8)
      s += Kg[(size_t)c * NN + rbase + rr] * xv[c];
    part[rr][l8] = s;
    __syncthreads();
    if (t < 32) {
      float s2 = 0.0f;
      for (int u = 0; u < 8; ++u) s2 += part[t][u];
      rhs[t] = xv[rbase + t] - s2;
    }
    __syncthreads();
    if (t < 32) { // wave-synchronous 32x32 upper (L^T) solve
      for (int jj = 31; jj >= 0; --jj) {
        if (t == jj) xv[rbase + jj] = rhs[jj] / Kg[(size_t)(rbase + jj) * NN + rbase + jj];
        if (t < jj) rhs[t] -= Kg[(size_t)(rbase + jj) * NN + rbase + t] * xv[rbase + jj];
      }
    }
    __syncthreads();
  }

  for (int i = t; i < NN; i += 256) Alpha[(size_t)g * NN + i] = xv[i];
}

// =====================================================================
// Stage 3a: det(cov_su @ diag(1/scales_g) + I)  via 40x40 LU with pivoting
// =====================================================================
__global__ void k_det(const float* __restrict__ cov_su, const float* __restrict__ scales,
                      float* __restrict__ detc) {
  const int g = blockIdx.x;
  const int t = threadIdx.x; // 64
  __shared__ float M[DD][DD + 1];
  __shared__ int pidx;
  __shared__ float detS;
  for (int idx = t; idx < DD * DD; idx += 64) {
    int r = idx / DD, c = idx % DD;
    M[r][c] = cov_su[r * DD + c] * (1.0f / scales[g * DD + c]) + ((r == c) ? 1.0f : 0.0f);
  }
  if (t == 0) detS = 1.0f;
  __syncthreads();

  for (int j = 0; j < DD; ++j) {
    if (t == 0) {
      int p = j;
      float mv = fabsf(M[j][j]);
      for (int i = j + 1; i < DD; ++i) {
        float a = fabsf(M[i][j]);
        if (a > mv) { mv = a; p = i; }
      }
      pidx = p;
      if (p != j) detS = -detS;
    }
    __syncthreads();
    const int p = pidx;
    if (p != j && t < DD) {
      float tmp = M[j][t]; M[j][t] = M[p][t]; M[p][t] = tmp;
    }
    __syncthreads();
    if (t == 0) detS *= M[j][j];
    const int i = j + 1 + t;
    if (i < DD) {
      float f = M[i][j] / M[j][j];
      for (int c = j; c < DD; ++c) M[i][c] -= f * M[j][c];
    }
    __syncthreads();
  }
  if (t == 0) detc[g] = detS;
}

// =====================================================================
// Stage 3b: Cholesky of (cov_su + diag(scales_g)), then per-point
// quadratic form ||L^{-1} nu_i||^2 and exp-quad weights.
// =====================================================================
__global__ void k_quad(const float* __restrict__ X, const float* __restrict__ mu,
                       const float* __restrict__ cov_su, const float* __restrict__ scales,
                       const float* __restrict__ detc, float* __restrict__ expq) {
  const int g = blockIdx.x;
  const int t = threadIdx.x; // 256
  __shared__ float Lm[DD][DD + 1];
  for (int idx = t; idx < DD * DD; idx += 256) {
    int r = idx / DD, c = idx % DD;
    Lm[r][c] = cov_su[r * DD + c] + ((r == c) ? scales[g * DD + c] : 0.0f);
  }
  __syncthreads();

  for (int j = 0; j < DD; ++j) {
    if (t == 0) Lm[j][j] = sqrtf(Lm[j][j]);
    __syncthreads();
    if (t > j && t < DD) Lm[t][j] = Lm[t][j] / Lm[j][j];
    __syncthreads();
    if (t > j && t < DD) {
      float f = Lm[t][j];
      for (int c = j + 1; c <= t; ++c) Lm[t][c] -= f * Lm[c][j];
    }
    __syncthreads();
  }

  const float dc = detc[g];
  for (int i = t; i < NN; i += 256) {
    float nu[DD];
#pragma unroll
    for (int d = 0; d < DD; ++d) nu[d] = X[(size_t)i * DD + d] - mu[d];
    float q = 0.0f;
#pragma unroll
    for (int j = 0; j < DD; ++j) {
      float v = nu[j];
#pragma unroll
      for (int c = 0; c < j; ++c) v -= Lm[j][c] * nu[c];
      v /= Lm[j][j];
      nu[j] = v;
      q += v * v;
    }
    expq[(size_t)g * NN + i] = dc * __expf(-0.5f * q);
  }
}

// =====================================================================
// Stage 4: mean[g] = dot(alpha_g, expq_g)
// =====================================================================
__global__ void k_dot(const float* __restrict__ Alpha, const float* __restrict__ expq,
                      float* __restrict__ out) {
  const int g = blockIdx.x;
  const int t = threadIdx.x; // 256
  float s = 0.0f;
  for (int i = t; i < NN; i += 256)
    s += Alpha[(size_t)g * NN + i] * expq[(size_t)g * NN + i];
  __shared__ float red[256];
  red[t] = s;
  __syncthreads();
  for (int o = 128; o > 0; o >>= 1) {
    if (t < o) red[t] += red[t + o];
    __syncthreads();
  }
  if (t == 0) out[g] = red[0];
}

// =====================================================================
extern "C" void kernel_launch(void* const* d_in, const int* in_sizes, int n_in,
                              void* d_out, int out_size, void* d_ws, size_t ws_size,
                              hipStream_t stream) {
  const float* X      = (const float*)d_in[0]; // [1024,40]
  const float* Y      = (const float*)d_in[1]; // [1024,32]
  const float* coeff  = (const float*)d_in[2]; // [32]
  const float* scales = (const float*)d_in[3]; // [32,40]
  const float* noise  = (const float*)d_in[4]; // [32]
  const float* mu     = (const float*)d_in[5]; // [40]
  const float* cov_su = (const float*)d_in[6]; // [40,40]
  float* out = (float*)d_out;                  // [32]

  char* ws = (char*)d_ws;
  float* Kmat  = (float*)ws;                                  // 32*1024*1024 f32 = 128 MB (fits 192 MB L2)
  float* LINV  = (float*)(ws + (size_t)GG * NN * NN * 4);     // 32*32*32
  float* ALPHA = LINV + (size_t)GG * 1024;                    // 32*1024
  float* EXPQ  = ALPHA + (size_t)GG * NN;                     // 32*1024
  float* DETC  = EXPQ + (size_t)GG * NN;                      // 32

  // Stage 0: build 32 kernel matrices
  k_build<<<dim3(NN / 64, NN / 64, GG), 256, 0, stream>>>(X, coeff, scales, noise, Kmat);

  // Stage 1: batched blocked Cholesky (panel width 32, WMMA SYRK/TRSM, TDM tile loads)
  for (int kp = 0; kp < NPAN; ++kp) {
    k_potf2<<<GG, 32, 0, stream>>>(Kmat, LINV, kp);
    const int m = NPAN - 1 - kp; // remaining 32-row tiles below panel
    if (m > 0) {
      k_trsm<<<dim3(m, GG), 128, 0, stream>>>(Kmat, LINV, kp);
      const int pairs = m * (m + 1) / 2;
      k_syrk<<<dim3(pairs, GG), 128, 0, stream>>>(Kmat, kp);
    }
  }

  // Stage 2: alpha_g = K_g^{-1} y_g
  k_cholsolve<<<GG, 256, 0, stream>>>(Kmat, Y, ALPHA);

  // Stage 3: small 40x40 det + quadratic-form weights
  k_det<<<GG, 64, 0, stream>>>(cov_su, scales, DETC);
  k_quad<<<GG, 256, 0, stream>>>(X, mu, cov_su, scales, DETC, EXPQ);

  // Stage 4: final dots
  k_dot<<<GG, 256, 0, stream>>>(ALPHA, EXPQ, out);
}